// P2L_Inter_Layer_21131239096468
// MI455X (gfx1250) — compile-verified
//
#include <hip/hip_runtime.h>
#include <hip/hip_bf16.h>
#include <math.h>

#define CZ     128
#define HD     120
#define NGAUSS 32
#define NHEAD  12
#define DHEAD  10
#define KVLD   164   // s_kv row stride (floats): 16B-aligned rows, conflict-free mod 64

typedef __attribute__((ext_vector_type(2))) float v2f;
typedef __attribute__((ext_vector_type(8))) float v8f;

__device__ __forceinline__ v8f wmma4(v2f a, v2f b, v8f c) {
  // D = A(16x4 f32) * B(4x16 f32) + C(16x16 f32)   -> v_wmma_f32_16x16x4_f32
  return __builtin_amdgcn_wmma_f32_16x16x4_f32(false, a, false, b, (short)0, c,
                                               false, false);
}

// One wave computes MT stacked 16x16 f32 tiles sharing the same B columns:
//   acc[mt] += A(mt-th 16xK tile, LDS, stride lda) * B(KbxNb global row-major)
// K multiple of 4; A zero-padded past real K; B row index clamped to kmaxb-1
// (clamped rows multiply zeros). ncol: per-lane, pre-clamped B column.
template <int MT>
__device__ __forceinline__ void gemm_tiles(const float* As, int lda,
                                           const float* __restrict__ Bg, int ldb,
                                           int kmaxb, int ncol, int K, int lane,
                                           v8f* acc) {
  const int m  = lane & 15;
  const int kh = (lane >> 4) << 1;   // lanes 0-15: K+0/K+1 ; lanes 16-31: K+2/K+3
  const float* Bp = Bg + ncol;
  for (int k = 0; k < K; k += 4) {
    const int kk = k + kh;
    const int k0 = (kk     < kmaxb) ? kk     : (kmaxb - 1);
    const int k1 = (kk + 1 < kmaxb) ? kk + 1 : (kmaxb - 1);
    v2f b;
    b.x = Bp[(size_t)k0 * ldb];
    b.y = Bp[(size_t)k1 * ldb];
#pragma unroll
    for (int mt = 0; mt < MT; ++mt) {
      v2f a;
      a.x = As[(mt * 16 + m) * lda + kk];
      a.y = As[(mt * 16 + m) * lda + kk + 1];
      acc[mt] = wmma4(a, b, acc[mt]);
    }
  }
}

template <int MT>
__device__ __forceinline__ void zero_acc(v8f* acc) {
#pragma unroll
  for (int mt = 0; mt < MT; ++mt)
#pragma unroll
    for (int i = 0; i < 8; ++i) acc[mt][i] = 0.f;
}

// Raw LDS byte offset of a __shared__ object (addrspace(3) cast).
__device__ __forceinline__ unsigned lds_offset(const void* gp) {
  return (unsigned)(unsigned long long)(__attribute__((address_space(3))) const void*)gp;
}

// Async memory->LDS copy of 16 bytes per lane (no VGPR round-trip, ASYNCcnt).
__device__ __forceinline__ void async_load_b128(unsigned lds_byte_off,
                                                const void* gaddr) {
  asm volatile("global_load_async_to_lds_b128 %0, %1, off"
               :: "v"(lds_byte_off), "v"(gaddr) : "memory");
}
__device__ __forceinline__ void wait_asynccnt0() {
  asm volatile("s_wait_asynccnt 0" ::: "memory");
}

// Order-preserving encode of f32 into u32 for atomicMax-based segment max.
__device__ __forceinline__ unsigned fenc(float f) {
  unsigned u = __float_as_uint(f);
  return (u & 0x80000000u) ? ~u : (u | 0x80000000u);
}
__device__ __forceinline__ float fdec(unsigned u) {
  unsigned b = (u & 0x80000000u) ? (u & 0x7FFFFFFFu) : ~u;
  return __uint_as_float(b);
}

// ---------------- Kernel 0: q = h @ Wq + bq  ([N,128] -> [N,120]) ----------------
// 64 nodes/block, 4 row-tiles per wave -> B fragments amortized 4x.
__global__ __launch_bounds__(256)
void q_proj_kernel(const float* __restrict__ h, const float* __restrict__ Wq,
                   const float* __restrict__ bq, float* __restrict__ qout, int N) {
  __shared__ float s_h[64][133];
  const int t  = threadIdx.x;
  const int n0 = blockIdx.x * 64;
#pragma unroll
  for (int i = 0; i < 32; ++i) {
    int f = i * 256 + t;
    int r = f >> 7, c = f & 127;
    int node = n0 + r; if (node >= N) node = N - 1;
    s_h[r][c] = h[(size_t)node * CZ + c];
  }
  __syncthreads();
  const int lane = t & 31, w = t >> 5;
  const int ncr = w * 16 + (lane & 15);           // 0..127 (8 waves x 16 cols)
  const int nc  = (ncr < HD) ? ncr : (HD - 1);    // clamp into Wq's 120 columns
  v8f acc[4];
  zero_acc<4>(acc);
  gemm_tiles<4>(&s_h[0][0], 133, Wq, HD, CZ, nc, CZ, lane, acc);
#pragma unroll
  for (int mt = 0; mt < 4; ++mt)
#pragma unroll
    for (int i = 0; i < 8; ++i) {
      int r = mt * 16 + i + ((lane >> 4) << 3);
      int node = n0 + r;
      if (node < N && ncr < HD)
        qout[(size_t)node * HD + ncr] = acc[mt][i] + bq[ncr];
    }
}

// -------- Kernel 1: per-edge RBF/edge_emb + K/V projections + logits + seg-max --------
// 32 edges/block, 2 row-tiles per wave. p-rows gathered with async-to-LDS B128,
// overlapped with the RBF exp computation (ASYNCcnt).
__global__ __launch_bounds__(256)
void edge_kv_kernel(const int* __restrict__ eidx, const float* __restrict__ pos,
                    const float* __restrict__ pos_p, const float* __restrict__ p,
                    const float* __restrict__ W_edge, const float* __restrict__ b_edge,
                    const float* __restrict__ Wk, const float* __restrict__ bk,
                    const float* __restrict__ Wv, const float* __restrict__ bv,
                    const float* __restrict__ qin,
                    float* __restrict__ vout, float* __restrict__ logits,
                    unsigned* __restrict__ menc, int E) {
  __shared__ int   s_row[32], s_col[32];
  __shared__ float s_d2[32];
  __shared__ float s_rbf[32][NGAUSS];
  __shared__ float s_kv[32][KVLD];  // [p[col](128) | edge_attr(32)], 16B-aligned rows
  __shared__ float s_k[32][129];    // K projection result (for logits)
  const int t  = threadIdx.x;
  const int e0 = blockIdx.x * 32;

  if (t < 32) {
    int eg = e0 + t; if (eg >= E) eg = E - 1;
    const int r = eidx[eg];
    const int c = eidx[(size_t)E + eg];
    s_row[t] = r; s_col[t] = c;
    float dx = pos[r * 3 + 0] - pos_p[c * 3 + 0];
    float dy = pos[r * 3 + 1] - pos_p[c * 3 + 1];
    float dz = pos[r * 3 + 2] - pos_p[c * 3 + 2];
    s_d2[t] = dx * dx + dy * dy + dz * dz;
  }
  __syncthreads();

  // Issue async gather of p[col] rows directly into LDS (32 rows x 512B).
  // Each lane moves one 16B chunk; both addresses 16B-aligned.
#pragma unroll
  for (int i = 0; i < 4; ++i) {
    int f = i * 256 + t;                // 16B chunk id in [0,1024)
    int e = f >> 5, cc = (f & 31) * 4;  // edge, float column (mult of 4)
    const float* gp = p + (size_t)s_col[e] * CZ + cc;
    async_load_b128(lds_offset(&s_kv[e][cc]), gp);
  }

  {
    const float delta = 30.0f / 31.0f;
    const float coeff = -0.5f / (delta * delta);
#pragma unroll
    for (int i = 0; i < 4; ++i) {                 // 32 edges x 32 gaussians
      int f = i * 256 + t;
      int e = f >> 5, g = f & 31;
      float d = s_d2[e] - delta * (float)g;
      s_rbf[e][g] = __expf(coeff * d * d);
    }
  }
  __syncthreads();

#pragma unroll
  for (int i = 0; i < 4; ++i) {                   // edge_attr = rbf @ W_edge + b_edge
    int f = i * 256 + t;
    int e = f >> 5, o = f & 31;
    float acc = b_edge[o];
#pragma unroll
    for (int g = 0; g < NGAUSS; ++g) acc += s_rbf[e][g] * W_edge[g * NGAUSS + o];
    s_kv[e][CZ + o] = acc;
  }
  wait_asynccnt0();                               // p-gather landed in LDS
  __syncthreads();

  const int lane = t & 31, w = t >> 5;
  const int ncr = w * 16 + (lane & 15);
  const int nc  = (ncr < HD) ? ncr : (HD - 1);

  { // K projection: [32,160] @ [160,120]
    v8f acc[2];
    zero_acc<2>(acc);
    gemm_tiles<2>(&s_kv[0][0], KVLD, Wk, HD, 160, nc, 160, lane, acc);
#pragma unroll
    for (int mt = 0; mt < 2; ++mt)
#pragma unroll
      for (int i = 0; i < 8; ++i) {
        int r = mt * 16 + i + ((lane >> 4) << 3);
        if (ncr < HD) s_k[r][ncr] = acc[mt][i] + bk[ncr];
      }
  }
  { // V projection: [32,160] @ [160,120] -> global scratch
    v8f acc[2];
    zero_acc<2>(acc);
    gemm_tiles<2>(&s_kv[0][0], KVLD, Wv, HD, 160, nc, 160, lane, acc);
#pragma unroll
    for (int mt = 0; mt < 2; ++mt)
#pragma unroll
      for (int i = 0; i < 8; ++i) {
        int r = mt * 16 + i + ((lane >> 4) << 3);
        int eg = e0 + r;
        if (eg < E && ncr < HD) vout[(size_t)eg * HD + ncr] = acc[mt][i] + bv[ncr];
      }
  }
  __syncthreads();

#pragma unroll
  for (int i = 0; i < 2; ++i) {                   // logits + segment max (32x12)
    int f = i * 256 + t;
    if (f < 32 * NHEAD) {
      int e = f / NHEAD, hh = f % NHEAD;
      int row = s_row[e];
      float acc = 0.f;
#pragma unroll
      for (int d = 0; d < DHEAD; ++d)
        acc += qin[(size_t)row * HD + hh * DHEAD + d] * s_k[e][hh * DHEAD + d];
      float l = acc * 0.3162277660168379f;        // 1/sqrt(10)
      int eg = e0 + e;
      if (eg < E) {
        logits[(size_t)eg * NHEAD + hh] = l;
        atomicMax(&menc[row * NHEAD + hh], fenc(l));
      }
    }
  }
}

// -------- Kernel 2: ex = exp(l - m); den += ex; agg += ex * v (segment sums) --------
__global__ __launch_bounds__(256)
void softmax_agg_kernel(const int* __restrict__ eidx, const float* __restrict__ logits,
                        const unsigned* __restrict__ menc, const float* __restrict__ vin,
                        float* __restrict__ den, float* __restrict__ agg, int E) {
  int e = blockIdx.x * 256 + threadIdx.x;
  if (e >= E) return;
  int row = eidx[e];
  float ex[NHEAD];
#pragma unroll
  for (int hh = 0; hh < NHEAD; ++hh) {
    float m = fdec(menc[row * NHEAD + hh]);
    ex[hh] = __expf(logits[(size_t)e * NHEAD + hh] - m);
    atomicAdd(&den[row * NHEAD + hh], ex[hh]);
  }
#pragma unroll
  for (int hh = 0; hh < NHEAD; ++hh) {
#pragma unroll
    for (int d = 0; d < DHEAD; ++d) {
      int j = hh * DHEAD + d;
      atomicAdd(&agg[(size_t)row * HD + j], ex[hh] * vin[(size_t)e * HD + j]);
    }
  }
}

// -------- Kernel 3: t=agg/den; out1=h+t@Wo+bo; out=silu(out1@W1+b1)@W2+b2 --------
// 32 nodes/block; t-buffer and y-buffer share LDS (disjoint lifetimes).
__global__ __launch_bounds__(256)
void node_out_kernel(const float* __restrict__ h, const float* __restrict__ den,
                     const float* __restrict__ agg,
                     const float* __restrict__ Wo, const float* __restrict__ bo,
                     const float* __restrict__ W1, const float* __restrict__ b1,
                     const float* __restrict__ W2, const float* __restrict__ b2,
                     float* __restrict__ out, int N) {
  // region0: 32x257 floats (y; first 32x133 used as t earlier)
  // region1: 32x133 floats (out1)
  __shared__ float smem[32 * 257 + 32 * 133];
  float* s_ty = smem;               // t: stride 133 / y: stride 257
  float* s_a  = smem + 32 * 257;    // out1: stride 133
  const int t  = threadIdx.x;
  const int n0 = blockIdx.x * 32;
#pragma unroll
  for (int i = 0; i < 16; ++i) {
    int f = i * 256 + t;
    int r = f >> 7, c = f & 127;
    int node = n0 + r; if (node >= N) node = N - 1;
    float val = 0.f;
    if (c < HD) {
      float dd = den[node * NHEAD + c / DHEAD];
      val = agg[(size_t)node * HD + c] / (dd + 1e-16f);
    }
    s_ty[r * 133 + c] = val;        // t, K padded 120->128 with zeros
  }
  __syncthreads();
  const int lane = t & 31, w = t >> 5;
  const int hi8 = (lane >> 4) << 3;
  { // GEMM1: [32,120(pad128)] @ Wo[120,128], residual + bias -> s_a
    int nc = w * 16 + (lane & 15);
    v8f acc[2];
    zero_acc<2>(acc);
    gemm_tiles<2>(s_ty, 133, Wo, CZ, HD, nc, CZ, lane, acc);
#pragma unroll
    for (int mt = 0; mt < 2; ++mt)
#pragma unroll
      for (int i = 0; i < 8; ++i) {
        int r = mt * 16 + i + hi8;
        int node = n0 + r; if (node >= N) node = N - 1;
        s_a[r * 133 + nc] = acc[mt][i] + h[(size_t)node * CZ + nc] + bo[nc];
      }
  }
  __syncthreads();                  // t dead; region0 becomes y
#pragma unroll
  for (int rep = 0; rep < 2; ++rep) { // GEMM2: [32,128] @ W1[128,256] -> silu -> y
    int nc = (w + 8 * rep) * 16 + (lane & 15);
    v8f acc[2];
    zero_acc<2>(acc);
    gemm_tiles<2>(s_a, 133, W1, 2 * CZ, CZ, nc, CZ, lane, acc);
#pragma unroll
    for (int mt = 0; mt < 2; ++mt)
#pragma unroll
      for (int i = 0; i < 8; ++i) {
        int r = mt * 16 + i + hi8;
        float x = acc[mt][i] + b1[nc];
        s_ty[r * 257 + nc] = x / (1.f + __expf(-x));
      }
  }
  __syncthreads();
  { // GEMM3: [32,256] @ W2[256,128] + b2 -> d_out
    int nc = w * 16 + (lane & 15);
    v8f acc[2];
    zero_acc<2>(acc);
    gemm_tiles<2>(s_ty, 257, W2, CZ, 2 * CZ, nc, 2 * CZ, lane, acc);
#pragma unroll
    for (int mt = 0; mt < 2; ++mt)
#pragma unroll
      for (int i = 0; i < 8; ++i) {
        int r = mt * 16 + i + hi8;
        int node = n0 + r;
        if (node < N) out[(size_t)node * CZ + nc] = acc[mt][i] + b2[nc];
      }
  }
}

extern "C" void kernel_launch(void* const* d_in, const int* in_sizes, int n_in,
                              void* d_out, int out_size, void* d_ws, size_t ws_size,
                              hipStream_t stream) {
  const float* pos    = (const float*)d_in[0];
  const float* h      = (const float*)d_in[1];
  const int*   eidx   = (const int*)  d_in[2];
  const float* pos_p  = (const float*)d_in[3];
  const float* p      = (const float*)d_in[4];
  const float* W_edge = (const float*)d_in[5];
  const float* b_edge = (const float*)d_in[6];
  const float* Wq     = (const float*)d_in[7];
  const float* bq     = (const float*)d_in[8];
  const float* Wk     = (const float*)d_in[9];
  const float* bk     = (const float*)d_in[10];
  const float* Wv     = (const float*)d_in[11];
  const float* bv     = (const float*)d_in[12];
  const float* Wo     = (const float*)d_in[13];
  const float* bo     = (const float*)d_in[14];
  const float* W1     = (const float*)d_in[15];
  const float* b1     = (const float*)d_in[16];
  const float* W2     = (const float*)d_in[17];
  const float* b2     = (const float*)d_in[18];
  float* out = (float*)d_out;

  const int N = in_sizes[1] / CZ;
  const int E = in_sizes[2] / 2;

  // workspace layout (floats)
  float*    q    = (float*)d_ws;                         // N*120
  float*    vbuf = q + (size_t)N * HD;                   // E*120
  float*    lg   = vbuf + (size_t)E * HD;                // E*12
  unsigned* menc = (unsigned*)(lg + (size_t)E * NHEAD);  // N*12
  float*    den  = (float*)(menc + (size_t)N * NHEAD);   // N*12
  float*    agg  = den + (size_t)N * NHEAD;              // N*120

  // zero menc/den/agg (contiguous block) — capture-safe async memset
  hipMemsetAsync(menc, 0,
                 ((size_t)N * NHEAD * 2 + (size_t)N * HD) * sizeof(float),
                 stream);

  const dim3 blk(256);
  const int nb_q       = (N + 63) / 64;
  const int nb_edge32  = (E + 31) / 32;
  const int nb_edge256 = (E + 255) / 256;
  const int nb_node32  = (N + 31) / 32;

  q_proj_kernel<<<nb_q, blk, 0, stream>>>(h, Wq, bq, q, N);
  edge_kv_kernel<<<nb_edge32, blk, 0, stream>>>(eidx, pos, pos_p, p, W_edge, b_edge,
                                                Wk, bk, Wv, bv, q, vbuf, lg, menc, E);
  softmax_agg_kernel<<<nb_edge256, blk, 0, stream>>>(eidx, lg, menc, vbuf, den, agg, E);
  node_out_kernel<<<nb_node32, blk, 0, stream>>>(h, den, agg, Wo, bo, W1, b1, W2, b2,
                                                 out, N);
}